// RGCNModel_41549513622112
// MI455X (gfx1250) — compile-verified
//
#include <hip/hip_runtime.h>

typedef __attribute__((ext_vector_type(2))) float v2f;
typedef __attribute__((ext_vector_type(8))) float v8f;

#define N_NODES  100000
#define N_EDGES  800000
#define H_DIM    64
#define OUT_DIM  256
#define NUM_RELS 8
#define PERM_CAP (N_EDGES + NUM_RELS * 16)   // 800128, divisible by 16
#define LDS_STRIDE 68                        // 64 + 4 pad: bank = (row*4+col)%64, conflict-free

// ---------------- counting sort of edges by relation ----------------

__global__ void rgcn_sort_init(int* __restrict__ counts, int* __restrict__ perm) {
    int i = blockIdx.x * blockDim.x + threadIdx.x;
    if (i < NUM_RELS) counts[i] = 0;
    if (i < PERM_CAP) perm[i] = -1;
}

__global__ __launch_bounds__(256) void rgcn_hist(const int* __restrict__ etypes,
                                                 int* __restrict__ counts) {
    __shared__ int lc[NUM_RELS];
    if (threadIdx.x < NUM_RELS) lc[threadIdx.x] = 0;
    __syncthreads();
    int e = blockIdx.x * blockDim.x + threadIdx.x;
    if (e < N_EDGES) atomicAdd(&lc[etypes[e]], 1);
    __syncthreads();
    if (threadIdx.x < NUM_RELS && lc[threadIdx.x]) atomicAdd(&counts[threadIdx.x], lc[threadIdx.x]);
}

__global__ void rgcn_scan(const int* __restrict__ counts, int* __restrict__ padOff,
                          int* __restrict__ cursor) {
    if (threadIdx.x == 0 && blockIdx.x == 0) {
        int acc = 0;
        for (int r = 0; r < NUM_RELS; ++r) {
            padOff[r] = acc;
            cursor[r] = acc;
            acc += ((counts[r] + 15) >> 4) << 4;   // 16-align each relation segment
        }
        padOff[NUM_RELS] = acc;
    }
}

__global__ __launch_bounds__(256) void rgcn_scatter(const int* __restrict__ etypes,
                                                    int* __restrict__ cursor,
                                                    int* __restrict__ perm) {
    __shared__ int lcount[NUM_RELS];
    __shared__ int lbase[NUM_RELS];
    if (threadIdx.x < NUM_RELS) lcount[threadIdx.x] = 0;
    __syncthreads();
    int e = blockIdx.x * blockDim.x + threadIdx.x;
    int t = 0, local = 0;
    bool ok = (e < N_EDGES);
    if (ok) {
        t = etypes[e];
        local = atomicAdd(&lcount[t], 1);
    }
    __syncthreads();
    if (threadIdx.x < NUM_RELS)
        lbase[threadIdx.x] = lcount[threadIdx.x] ? atomicAdd(&cursor[threadIdx.x], lcount[threadIdx.x]) : 0;
    __syncthreads();
    if (ok) perm[lbase[t] + local] = e;
}

// ---------------- self-loop: out = h @ W_loop + bias ----------------
// One block = one 16-node tile. A staged in LDS once; 8 waves x 2 col-tiles each.

__global__ __launch_bounds__(256) void rgcn_selfloop(const float* __restrict__ h,
                                                     const float* __restrict__ Wl,
                                                     const float* __restrict__ bias,
                                                     float* __restrict__ out) {
    __shared__ float As[16 * LDS_STRIDE];

    const int nodeTile = blockIdx.x;
    {   // stage A: thread t loads 4 floats of row t/16
        const int row = threadIdx.x >> 4;
        const int seg = threadIdx.x & 15;
        const float4 av = *(const float4*)(h + (size_t)(nodeTile * 16 + row) * H_DIM + seg * 4);
        float* dA = &As[row * LDS_STRIDE + seg * 4];
        dA[0] = av.x; dA[1] = av.y; dA[2] = av.z; dA[3] = av.w;
    }
    __syncthreads();

    const int lane  = threadIdx.x & 31;
    const int wave  = threadIdx.x >> 5;
    const int myN   = lane & 15;
    const int khalf = (lane >> 4) * 2;
    const int colBase0 = (wave * 2 + 0) * 16;
    const int colBase1 = (wave * 2 + 1) * 16;
    const float* aLds = &As[(lane & 15) * LDS_STRIDE + khalf];

    v8f c0 = {}, c1 = {};
#pragma unroll
    for (int k0 = 0; k0 < H_DIM; k0 += 4) {
        v2f a, b0, b1;
        a.x = aLds[k0 + 0];
        a.y = aLds[k0 + 1];
        const float* wrow = Wl + (size_t)(k0 + khalf) * OUT_DIM;
        b0.x = wrow[colBase0 + myN];
        b0.y = wrow[OUT_DIM + colBase0 + myN];
        b1.x = wrow[colBase1 + myN];
        b1.y = wrow[OUT_DIM + colBase1 + myN];
        c0 = __builtin_amdgcn_wmma_f32_16x16x4_f32(false, a, false, b0, (short)0, c0, false, false);
        c1 = __builtin_amdgcn_wmma_f32_16x16x4_f32(false, a, false, b1, (short)0, c1, false, false);
    }

    const float bv0 = bias[colBase0 + myN];
    const float bv1 = bias[colBase1 + myN];
    const int rbase = (lane >> 4) * 8;
#pragma unroll
    for (int i = 0; i < 8; ++i) {
        int m = nodeTile * 16 + rbase + i;
        out[(size_t)m * OUT_DIM + colBase0 + myN] = c0[i] + bv0;
        out[(size_t)m * OUT_DIM + colBase1 + myN] = c1[i] + bv1;
    }
}

// ---------------- edge messages: out[dst] += (norm*h[src]) @ W[rel] ----------------
// One block = one 16-edge tile (single relation, segments are 16-aligned).
// Norm-scaled gathered A staged in LDS once; 8 waves x 2 col-tiles each.

__global__ __launch_bounds__(256) void rgcn_edge_msg(const float* __restrict__ h,
                                                     const float* __restrict__ W,
                                                     const float* __restrict__ norm,
                                                     const int* __restrict__ src,
                                                     const int* __restrict__ dst,
                                                     const int* __restrict__ perm,
                                                     const int* __restrict__ padOff,
                                                     float* __restrict__ out) {
    __shared__ float As[16 * LDS_STRIDE];
    __shared__ int   dstS[16];

    const int tileStart = blockIdx.x * 16;
    if (tileStart >= padOff[NUM_RELS]) return;       // uniform across block

    int rel = 0;
#pragma unroll
    for (int r = 1; r < NUM_RELS; ++r)
        if (tileStart >= padOff[r]) rel = r;         // segments are 16-aligned

    {   // stage norm-scaled A rows; padding rows (-1) become zero rows
        const int row = threadIdx.x >> 4;
        const int seg = threadIdx.x & 15;
        const int e   = perm[tileStart + row];
        const bool valid = (e >= 0);
        const float nrm  = valid ? norm[e] : 0.0f;
        const float* arow = h + (size_t)(valid ? src[e] : 0) * H_DIM;
        const float4 av = *(const float4*)(arow + seg * 4);
        float* dA = &As[row * LDS_STRIDE + seg * 4];
        dA[0] = nrm * av.x; dA[1] = nrm * av.y; dA[2] = nrm * av.z; dA[3] = nrm * av.w;
        if (seg == 0) dstS[row] = valid ? dst[e] : -1;
    }
    __syncthreads();

    const int lane  = threadIdx.x & 31;
    const int wave  = threadIdx.x >> 5;
    const int myN   = lane & 15;
    const int khalf = (lane >> 4) * 2;
    const int colBase0 = (wave * 2 + 0) * 16;
    const int colBase1 = (wave * 2 + 1) * 16;
    const float* Wr   = W + (size_t)rel * H_DIM * OUT_DIM;
    const float* aLds = &As[(lane & 15) * LDS_STRIDE + khalf];

    v8f c0 = {}, c1 = {};
#pragma unroll
    for (int k0 = 0; k0 < H_DIM; k0 += 4) {
        v2f a, b0, b1;
        a.x = aLds[k0 + 0];
        a.y = aLds[k0 + 1];
        const float* wrow = Wr + (size_t)(k0 + khalf) * OUT_DIM;
        b0.x = wrow[colBase0 + myN];
        b0.y = wrow[OUT_DIM + colBase0 + myN];
        b1.x = wrow[colBase1 + myN];
        b1.y = wrow[OUT_DIM + colBase1 + myN];
        c0 = __builtin_amdgcn_wmma_f32_16x16x4_f32(false, a, false, b0, (short)0, c0, false, false);
        c1 = __builtin_amdgcn_wmma_f32_16x16x4_f32(false, a, false, b1, (short)0, c1, false, false);
    }

    const int rbase = (lane >> 4) * 8;
#pragma unroll
    for (int i = 0; i < 8; ++i) {
        int d = dstS[rbase + i];
        if (d >= 0) {
            unsafeAtomicAdd(&out[(size_t)d * OUT_DIM + colBase0 + myN], c0[i]);
            unsafeAtomicAdd(&out[(size_t)d * OUT_DIM + colBase1 + myN], c1[i]);
        }
    }
}

// ---------------- relu epilogue ----------------

__global__ void rgcn_relu(float* __restrict__ out, int n) {
    int i = blockIdx.x * blockDim.x + threadIdx.x;
    if (i < n) out[i] = fmaxf(out[i], 0.0f);
}

// ---------------- launch ----------------

extern "C" void kernel_launch(void* const* d_in, const int* in_sizes, int n_in,
                              void* d_out, int out_size, void* d_ws, size_t ws_size,
                              hipStream_t stream) {
    const float* h    = (const float*)d_in[0];
    const float* W    = (const float*)d_in[1];
    const float* Wl   = (const float*)d_in[2];
    const float* bias = (const float*)d_in[3];
    const float* nrm  = (const float*)d_in[4];
    const int*   src  = (const int*)d_in[5];
    const int*   dst  = (const int*)d_in[6];
    const int*   et   = (const int*)d_in[7];
    float* out = (float*)d_out;

    int* wsI    = (int*)d_ws;
    int* counts = wsI;            // 8
    int* padOff = wsI + 8;        // 9
    int* cursor = wsI + 17;       // 8
    int* perm   = wsI + 32;       // PERM_CAP

    rgcn_sort_init<<<(PERM_CAP + 255) / 256, 256, 0, stream>>>(counts, perm);
    rgcn_hist<<<(N_EDGES + 255) / 256, 256, 0, stream>>>(et, counts);
    rgcn_scan<<<1, 32, 0, stream>>>(counts, padOff, cursor);
    rgcn_scatter<<<(N_EDGES + 255) / 256, 256, 0, stream>>>(et, cursor, perm);

    rgcn_selfloop<<<N_NODES / 16, 256, 0, stream>>>(h, Wl, bias, out);
    rgcn_edge_msg<<<PERM_CAP / 16, 256, 0, stream>>>(h, W, nrm, src, dst, perm, padOff, out);
    rgcn_relu<<<(N_NODES * OUT_DIM + 255) / 256, 256, 0, stream>>>(out, N_NODES * OUT_DIM);
}